// Model_HY_86371792322834
// MI455X (gfx1250) — compile-verified
//
#include <hip/hip_runtime.h>
#include <hip/hip_bf16.h>

#define EPSV 1e-5f

typedef float v2f __attribute__((ext_vector_type(2)));
typedef float v8f __attribute__((ext_vector_type(8)));

// ---------------------------------------------------------------- utilities
__global__ void zero_f32(float* __restrict__ p, long n) {
    long t = (long)blockIdx.x * blockDim.x + threadIdx.x;
    if (t < n) p[t] = 0.0f;
}

// ------------------------------------------------------------- GraphNorm
// stats[0:32] = column sums, stats[32:64] = column sum-of-squares
__global__ void gn_stats(const int* __restrict__ xidx, const float* __restrict__ emb,
                         float* __restrict__ stats, long N) {
    __shared__ float ssum[32];
    __shared__ float ssq[32];
    if (threadIdx.x < 32) { ssum[threadIdx.x] = 0.0f; ssq[threadIdx.x] = 0.0f; }
    __syncthreads();
    const int  c      = threadIdx.x & 31;
    long       i      = ((long)blockIdx.x * blockDim.x + threadIdx.x) >> 5;
    const long stride = ((long)gridDim.x * blockDim.x) >> 5;
    float s = 0.0f, q = 0.0f;
    for (; i < N; i += stride) {
        float v = emb[(long)xidx[i] * 32 + c];
        s += v; q += v * v;
    }
    atomicAdd(&ssum[c], s);
    atomicAdd(&ssq[c], q);
    __syncthreads();
    if (threadIdx.x < 32) {
        atomicAdd(&stats[threadIdx.x],      ssum[threadIdx.x]);
        atomicAdd(&stats[32 + threadIdx.x], ssq[threadIdx.x]);
    }
}

// var(x - ms*mean) = E[x^2] - (2*ms - ms^2) * mean^2
__global__ void gn_finalize(const float* __restrict__ stats, const float* __restrict__ w,
                            const float* __restrict__ b, const float* __restrict__ ms,
                            float invN, float* __restrict__ scale, float* __restrict__ shift) {
    int c = threadIdx.x;
    float mean = stats[c] * invN;
    float eh2  = stats[32 + c] * invN;
    float m    = ms[c];
    float var  = eh2 - (2.0f * m - m * m) * mean * mean;
    float sc   = w[c] / sqrtf(var + EPSV);
    scale[c] = sc;
    shift[c] = b[c] - sc * m * mean;
}

__global__ void gn_apply(const int* __restrict__ xidx, const float* __restrict__ emb,
                         const float* __restrict__ scale, const float* __restrict__ shift,
                         float* __restrict__ h, long N) {
    long t = (long)blockIdx.x * blockDim.x + threadIdx.x;
    if (t >= N * 32) return;
    long i = t >> 5;
    int  c = (int)(t & 31);
    h[t] = scale[c] * emb[(long)xidx[i] * 32 + c] + shift[c];
}

// --------------------------------------------------- edge scatter-add (mean agg)
// one thread per (edge, float4 chunk); chunk 0 also bumps the in-degree count
__global__ void scatter_add(const int* __restrict__ src, const int* __restrict__ dst,
                            long E, int dchunks, int d, const float* __restrict__ x,
                            float* __restrict__ agg, float* __restrict__ cnt) {
    long t = (long)blockIdx.x * blockDim.x + threadIdx.x;
    long total = E * (long)dchunks;
    if (t >= total) return;
    long e  = t / dchunks;
    int  ch = (int)(t - e * (long)dchunks);
    int  s  = src[e];
    int  dn = dst[e];
    const float4 v = *(const float4*)(x + (long)s * d + ch * 4);
    float* base = agg + (long)dn * d + ch * 4;
    atomicAdd(base + 0, v.x);
    atomicAdd(base + 1, v.y);
    atomicAdd(base + 2, v.z);
    atomicAdd(base + 3, v.w);
    if (ch == 0) atomicAdd(cnt + dn, 1.0f);
}

// -------------------------------------------- fused SAGE linear via f32 WMMA
// out[i,:] = relu( (agg[i,:]/max(cnt,1)) @ Wl + bl + x[i,:] @ Wr ), 32 outputs.
// One wave -> 16 rows x 32 cols using V_WMMA_F32_16X16X4_F32, K looped in 4s.
// A 16x4 layout: lane (l&15)=row, VGPR0/1 hold K = 2*(l>>4), 2*(l>>4)+1.
// B 4x16 layout: lane (l&15)=col, same K-halving. C/D: v8f, row r+8*(l>>4), col l&15.
__global__ void sage_gemm_wmma(int ntiles, int K1, const float* __restrict__ agg,
                               const float* __restrict__ cnt, const float* __restrict__ x,
                               const float* __restrict__ Wl, const float* __restrict__ bl,
                               const float* __restrict__ Wr, float* __restrict__ out) {
    const int lane = threadIdx.x & 31;
    const int wave = threadIdx.x >> 5;
    const int tile = blockIdx.x * (blockDim.x >> 5) + wave;
    if (tile >= ntiles) return;              // wave-uniform: EXEC stays all-ones
    const int  m   = lane & 15;
    const int  kh  = lane >> 4;
    const long row = (long)tile * 16 + m;

    const float invc = 1.0f / fmaxf(cnt[row], 1.0f);
    const float* arow = agg + row * K1;
    const float* xrow = x   + row * K1;

    v8f c0 = {};
    v8f c1 = {};

    // half 1: normalized aggregate against Wl
#pragma unroll 4
    for (int k0 = 0; k0 < K1; k0 += 4) {
        const int ka = k0 + 2 * kh;
        v2f a; a.x = arow[ka] * invc; a.y = arow[ka + 1] * invc;
        v2f b0, b1;
        b0.x = Wl[ka * 32 + m];        b0.y = Wl[(ka + 1) * 32 + m];
        b1.x = Wl[ka * 32 + 16 + m];   b1.y = Wl[(ka + 1) * 32 + 16 + m];
        c0 = __builtin_amdgcn_wmma_f32_16x16x4_f32(false, a, false, b0, (short)0, c0, false, false);
        c1 = __builtin_amdgcn_wmma_f32_16x16x4_f32(false, a, false, b1, (short)0, c1, false, false);
    }
    // half 2: self features against Wr
#pragma unroll 4
    for (int k0 = 0; k0 < K1; k0 += 4) {
        const int ka = k0 + 2 * kh;
        v2f a; a.x = xrow[ka]; a.y = xrow[ka + 1];
        v2f b0, b1;
        b0.x = Wr[ka * 32 + m];        b0.y = Wr[(ka + 1) * 32 + m];
        b1.x = Wr[ka * 32 + 16 + m];   b1.y = Wr[(ka + 1) * 32 + 16 + m];
        c0 = __builtin_amdgcn_wmma_f32_16x16x4_f32(false, a, false, b0, (short)0, c0, false, false);
        c1 = __builtin_amdgcn_wmma_f32_16x16x4_f32(false, a, false, b1, (short)0, c1, false, false);
    }

    const float blo = bl[m];
    const float bhi = bl[16 + m];
#pragma unroll
    for (int r = 0; r < 8; ++r) {
        const long orow = (long)tile * 16 + r + 8 * kh;
        out[orow * 32 + m]      = fmaxf(c0[r] + blo, 0.0f);
        out[orow * 32 + 16 + m] = fmaxf(c1[r] + bhi, 0.0f);
    }
}

// ---------------------------------------------------------- pair gather [M,64]
__global__ void pair_gather(const int* __restrict__ pos1, const float* __restrict__ hsrc,
                            float* __restrict__ hdst, long M) {
    long t = (long)blockIdx.x * blockDim.x + threadIdx.x;
    if (t >= M * 64) return;
    long m = t >> 6;
    int  c = (int)(t & 63);
    int  s = pos1[m * 2 + (c >> 5)];
    hdst[t] = hsrc[(long)s * 32 + (c & 31)];
}

// ------------------------------------- final: concat(h[pos2[m0]],h[pos2[m1]])@W+b
// one wave32 per output row: lane l covers column l of both halves.
__global__ void pred_kernel(const float* __restrict__ h, const int* __restrict__ pos2,
                            const int* __restrict__ mask0, const int* __restrict__ mask1,
                            const float* __restrict__ W, const float* __restrict__ bias,
                            float* __restrict__ out, long Q) {
    long wid  = ((long)blockIdx.x * blockDim.x + threadIdx.x) >> 5;
    int  lane = threadIdx.x & 31;
    if (wid >= Q) return;                    // wave-uniform
    int i0 = pos2[mask0[wid]];
    int i1 = pos2[mask1[wid]];
    float v = h[(long)i0 * 32 + lane] * W[lane] +
              h[(long)i1 * 32 + lane] * W[32 + lane];
#pragma unroll
    for (int off = 16; off > 0; off >>= 1) v += __shfl_down(v, off, 32);
    if (lane == 0) out[wid] = v + bias[0];
}

// ------------------------------------------------------------------ launcher
static inline unsigned gblk(long n, int t) { return (unsigned)((n + t - 1) / t); }

extern "C" void kernel_launch(void* const* d_in, const int* in_sizes, int n_in,
                              void* d_out, int out_size, void* d_ws, size_t ws_size,
                              hipStream_t stream) {
    const int*   x     = (const int*)d_in[0];
    const int*   edge1 = (const int*)d_in[1];
    const int*   edge2 = (const int*)d_in[2];
    const int*   pos1  = (const int*)d_in[3];
    const int*   pos2  = (const int*)d_in[4];
    const int*   mask0 = (const int*)d_in[5];
    const int*   mask1 = (const int*)d_in[6];
    const float* emb   = (const float*)d_in[7];
    const float* gw    = (const float*)d_in[8];
    const float* gb    = (const float*)d_in[9];
    const float* gms   = (const float*)d_in[10];
    const float* c1Wl  = (const float*)d_in[11];
    const float* c1bl  = (const float*)d_in[12];
    const float* c1Wr  = (const float*)d_in[13];
    const float* c2aWl = (const float*)d_in[14];
    const float* c2abl = (const float*)d_in[15];
    const float* c2aWr = (const float*)d_in[16];
    const float* c2bWl = (const float*)d_in[17];
    const float* c2bbl = (const float*)d_in[18];
    const float* c2bWr = (const float*)d_in[19];
    const float* predW = (const float*)d_in[20];
    const float* predB = (const float*)d_in[21];
    float* out = (float*)d_out;

    const long N  = in_sizes[0];
    const long E1 = (long)in_sizes[1] / 2;
    const long E2 = (long)in_sizes[2] / 2;
    const long M  = (long)in_sizes[3] / 2;
    const long Q  = in_sizes[5];

    float* ws = (float*)d_ws;
    size_t o = 0;
    auto alloc = [&](size_t nf) { size_t r = o; o += (nf + 63) & ~(size_t)63; return r; };
    float* stats = ws + alloc(128);       // sums | sumsq | scale | shift
    float* hA    = ws + alloc((size_t)N * 32);
    float* hB    = ws + alloc((size_t)N * 32);
    float* aggN  = ws + alloc((size_t)N * 32);
    float* cntN  = ws + alloc((size_t)N);
    float* hM2   = ws + alloc((size_t)M * 64);
    float* hMa   = ws + alloc((size_t)M * 32);
    float* aggM  = ws + alloc((size_t)M * 64);
    float* cntM  = ws + alloc((size_t)M);
    float* hMb   = hM2;                   // hM2 dead after c2a -> reuse for c2b output
    (void)ws_size; (void)n_in; (void)out_size;

    const int T = 256;

    // ---- GraphNorm on gathered embeddings
    zero_f32<<<1, 64, 0, stream>>>(stats, 64);
    gn_stats<<<1024, T, 0, stream>>>(x, emb, stats, N);
    gn_finalize<<<1, 32, 0, stream>>>(stats, gw, gb, gms, 1.0f / (float)N,
                                      stats + 64, stats + 96);
    gn_apply<<<gblk(N * 32, T), T, 0, stream>>>(x, emb, stats + 64, stats + 96, hA, N);

    // ---- conv1 layer 0 (32->32) on edge1: hA -> hB
    zero_f32<<<gblk(N * 32, T), T, 0, stream>>>(aggN, N * 32);
    zero_f32<<<gblk(N, T), T, 0, stream>>>(cntN, N);
    scatter_add<<<gblk(E1 * 8, T), T, 0, stream>>>(edge1, edge1 + E1, E1, 8, 32, hA, aggN, cntN);
    sage_gemm_wmma<<<gblk(N / 16, 8), T, 0, stream>>>((int)(N / 16), 32, aggN, cntN, hA,
                                                      c1Wl, c1bl, c1Wr, hB);

    // ---- conv1 layer 1 (32->32) on edge1: hB -> hA
    zero_f32<<<gblk(N * 32, T), T, 0, stream>>>(aggN, N * 32);
    zero_f32<<<gblk(N, T), T, 0, stream>>>(cntN, N);
    scatter_add<<<gblk(E1 * 8, T), T, 0, stream>>>(edge1, edge1 + E1, E1, 8, 32, hB, aggN, cntN);
    sage_gemm_wmma<<<gblk(N / 16, 8), T, 0, stream>>>((int)(N / 16), 32, aggN, cntN, hB,
                                                      c1Wl + 1024, c1bl + 32, c1Wr + 1024, hA);

    // ---- pair gather: [M, 64]
    pair_gather<<<gblk(M * 64, T), T, 0, stream>>>(pos1, hA, hM2, M);

    // ---- conv2a (64->32) on edge2: hM2 -> hMa
    zero_f32<<<gblk(M * 64, T), T, 0, stream>>>(aggM, M * 64);
    zero_f32<<<gblk(M, T), T, 0, stream>>>(cntM, M);
    scatter_add<<<gblk(E2 * 16, T), T, 0, stream>>>(edge2, edge2 + E2, E2, 16, 64, hM2, aggM, cntM);
    sage_gemm_wmma<<<gblk(M / 16, 8), T, 0, stream>>>((int)(M / 16), 64, aggM, cntM, hM2,
                                                      c2aWl, c2abl, c2aWr, hMa);

    // ---- conv2b (32->32) on edge2: hMa -> hMb (reuses hM2 space)
    zero_f32<<<gblk(M * 32, T), T, 0, stream>>>(aggM, M * 32);
    zero_f32<<<gblk(M, T), T, 0, stream>>>(cntM, M);
    scatter_add<<<gblk(E2 * 8, T), T, 0, stream>>>(edge2, edge2 + E2, E2, 8, 32, hMa, aggM, cntM);
    sage_gemm_wmma<<<gblk(M / 16, 8), T, 0, stream>>>((int)(M / 16), 32, aggM, cntM, hMa,
                                                      c2bWl, c2bbl, c2bWr, hMb);

    // ---- final prediction: one wave per output row
    pred_kernel<<<gblk(Q * 32, T), T, 0, stream>>>(hMb, pos2, mask0, mask1, predW, predB, out, Q);
}